// LocallyConnectedLayer_72670846649181
// MI455X (gfx1250) — compile-verified
//
#include <hip/hip_runtime.h>
#include <stdint.h>

// ---------------- problem constants (from reference) ----------------
constexpr int Bn  = 64;               // batch
constexpr int Cn  = 64;               // in channels
constexpr int On  = 64;               // out channels
constexpr int Hn  = 32, Wn = 32;
constexpr int Ln  = Hn * Wn;          // 1024 pixels
constexpr int KS  = 3;                // 3x3 kernel
constexpr int CKK = Cn * KS * KS;     // 576 contraction length

// ---------------- tiling ----------------
// 1024 independent per-pixel GEMMs (M=64 batch, N=64 O, K=576, fp32).
// Workgroup: 4 consecutive pixels (l), one wave each -> weight global reads are
// 16B runs along l; 192MB L2 (weight tensor = 151MB fits) merges lines across
// adjacent-l workgroups and absorbs the 4x batch-chunk rescan.
constexpr int PX    = 4;              // pixels per workgroup
constexpr int NWAVE = PX;             // one wave per pixel
constexpr int NTHR  = NWAVE * 32;     // 128 threads
constexpr int BC    = 16;             // batch chunk = WMMA M
constexpr int CCH   = 4;              // channels per k-chunk
constexpr int KC    = CCH * KS * KS;  // 36 contraction elems / chunk (9 K=4 steps)
constexpr int XC    = PX + 2;         // x tile cols incl. halo = 6

typedef float        v2f  __attribute__((ext_vector_type(2)));
typedef float        v8f  __attribute__((ext_vector_type(8)));
typedef unsigned int u32x4 __attribute__((ext_vector_type(4)));
typedef int          i32x4 __attribute__((ext_vector_type(4)));
typedef int          i32x8 __attribute__((ext_vector_type(8)));

#if __has_include(<hip/amd_detail/amd_gfx1250_TDM.h>)
#define TDM_6ARG 1
#endif

__global__ __launch_bounds__(NTHR)
void lcl_wmma_kernel(const float* __restrict__ x,
                     const float* __restrict__ w,
                     const float* __restrict__ bias,
                     float* __restrict__ out)
{
  // TDM writes x-fastest -> layout [o][k][l]; 64*36*4*4B = 36KB
  __shared__ float lds_w[On][KC][PX];
  // padded input halo tile: [b][c][row][col]; 16*4*3*6*4B = 4.5KB
  __shared__ float lds_x[BC][CCH][KS][XC];

  const int tid  = threadIdx.x;
  const int lane = tid & 31;
  const int wv   = tid >> 5;              // wave id == pixel within block
  const int m    = lane & 15;             // M / N row within 16
  const int kh   = (lane >> 4) << 1;      // K half: lanes 0-15 -> K{0,1}, 16-31 -> K{2,3}

  const int l0 = blockIdx.x * PX;         // PX=4 never crosses an image row
  const int y0 = l0 >> 5;
  const int x0 = l0 & 31;
  const int p  = l0 + wv;                 // this wave's pixel

  const uint32_t ldsw_off =
      (uint32_t)(uintptr_t)(&lds_w[0][0][0]);  // flat LDS addr: low 32b = LDS byte offset

  // bias[o, l] — same value for every batch row of a tile
  float bv[4];
#pragma unroll
  for (int nt = 0; nt < 4; ++nt)
    bv[nt] = bias[(size_t)(nt * 16 + m) * Ln + p];

  for (int bc = 0; bc < Bn / BC; ++bc) {
    v8f acc[4];
#pragma unroll
    for (int nt = 0; nt < 4; ++nt)
#pragma unroll
      for (int i = 0; i < 8; ++i) acc[nt][i] = bv[nt];

    for (int cc = 0; cc < Cn / CCH; ++cc) {
      __syncthreads();  // previous chunk's LDS reads complete before overwrite

      // ---- TDM: weight[0:64][cc*KC : cc*KC+KC][l0 : l0+PX] -> lds_w --------
      // 3D tile: dim0 = l (contiguous, 4 elems), dim1 = k (stride Ln),
      //          dim2 = o (stride CKK*Ln). One issue per workgroup (wave 0).
      if (tid < 32) {
        const uint64_t ga = (uint64_t)(uintptr_t)w +
                            4ull * ((uint64_t)(cc * KC) * Ln + (uint64_t)l0);
        const uint64_t d1s = (uint64_t)CKK * Ln;  // 589824
        u32x4 g0; i32x8 g1; i32x4 g2, g3;
        g0[0] = 1u;                                   // count=1, user mode
        g0[1] = ldsw_off;                             // lds_addr
        g0[2] = (uint32_t)ga;                         // global_addr[31:0]
        g0[3] = (uint32_t)((ga >> 32) & 0x1FFFFFFu) | (2u << 30); // addr hi | type=2
        g1[0] = (int)(2u << 16);                      // data_size = 4B, no mask/flags
        g1[1] = (int)(((uint32_t)Ln & 0xFFFFu) << 16);        // tensor_dim0 lo16
        g1[2] = (int)((((uint32_t)Ln >> 16) & 0xFFFFu) |
                      (((uint32_t)CKK & 0xFFFFu) << 16));     // dim0 hi | dim1 lo
        g1[3] = (int)((((uint32_t)CKK >> 16) & 0xFFFFu) |
                      ((uint32_t)PX << 16));                  // dim1 hi | tile_dim0=4
        g1[4] = (int)(((uint32_t)On << 16) | (uint32_t)KC);   // tile_dim1=36 | tile_dim2=64
        g1[5] = (int)(uint32_t)Ln;                            // tensor_dim0_stride[31:0]
        g1[6] = (int)(((uint32_t)(d1s & 0xFFFFu)) << 16);     // d0s hi16=0 | d1s[15:0]
        g1[7] = (int)(uint32_t)(d1s >> 16);                   // tensor_dim1_stride[47:16]
        g2[0] = On;   // tensor_dim2
        g2[1] = 1;    // tensor_dim3 (tile_dim3=0 -> unused)
        g2[2] = 0;    // tensor_dim2_stride lo (unused)
        g2[3] = 0;    // stride hi | tile_dim3=0
        g3[0] = 0; g3[1] = 0; g3[2] = 0; g3[3] = 0;
#ifdef TDM_6ARG
        i32x8 gz = {};
        __builtin_amdgcn_tensor_load_to_lds(g0, g1, g2, g3, gz, 0);
#else
        __builtin_amdgcn_tensor_load_to_lds(g0, g1, g2, g3, 0);
#endif
      }

      // ---- cooperative x halo tile load (overlaps the TDM) ------------------
      {
        const int c0 = cc * CCH, b0 = bc * BC;
        for (int e = tid; e < BC * CCH * KS * XC; e += NTHR) {  // 1152/128 = 9 each
          int col = e % XC, t = e / XC;
          int r  = t % KS;  t /= KS;
          int ci = t % CCH;
          int bi = t / CCH;
          int yy = y0 - 1 + r;
          int xx = x0 - 1 + col;
          float v = 0.f;
          if (yy >= 0 && yy < Hn && xx >= 0 && xx < Wn)
            v = x[(((size_t)(b0 + bi)) * Cn + (c0 + ci)) * Ln + yy * Wn + xx];
          lds_x[bi][ci][r][col] = v;
        }
      }

      if (tid < 32) __builtin_amdgcn_s_wait_tensorcnt(0);
      __syncthreads();

      // ---- 9 WMMA K=4 steps over the 36-element chunk ----------------------
#pragma unroll
      for (int ks = 0; ks < KC / 4; ++ks) {
        const int k0 = ks * 4 + kh;       // this lane's K pair
        const int k1 = k0 + 1;
        // unfold index: k = c*9 + r*3 + j
        int c0i = k0 / 9, rr0 = k0 - c0i * 9, r0 = rr0 / 3, j0 = rr0 - r0 * 3;
        int c1i = k1 / 9, rr1 = k1 - c1i * 9, r1 = rr1 / 3, j1 = rr1 - r1 * 3;
        v2f a;
        a.x = lds_x[m][c0i][r0][wv + j0];  // A[m=batch][k0]
        a.y = lds_x[m][c1i][r1][wv + j1];  // A[m][k0+1]
#pragma unroll
        for (int nt = 0; nt < 4; ++nt) {
          v2f bf;
          bf.x = lds_w[nt * 16 + m][k0][wv];  // B[k0][n=o]
          bf.y = lds_w[nt * 16 + m][k1][wv];  // B[k0+1][n]
          acc[nt] = __builtin_amdgcn_wmma_f32_16x16x4_f32(
              false, a, false, bf, (short)0, acc[nt], false, false);
        }
      }
    }

    // ---- store 16x64 tile: D vgpr v -> M = v (+8 for upper lanes), N = lane&15
    const int bb = bc * BC + ((lane >> 4) << 3);
#pragma unroll
    for (int nt = 0; nt < 4; ++nt) {
      const int o = nt * 16 + m;
#pragma unroll
      for (int v = 0; v < 8; ++v)
        out[(((size_t)(bb + v)) * On + o) * Ln + p] = acc[nt][v];
    }
  }
}

extern "C" void kernel_launch(void* const* d_in, const int* in_sizes, int n_in,
                              void* d_out, int out_size, void* d_ws, size_t ws_size,
                              hipStream_t stream) {
  const float* x  = (const float*)d_in[0];
  const float* w  = (const float*)d_in[1];
  const float* bs = (const float*)d_in[2];
  float* out = (float*)d_out;
  dim3 grid(Ln / PX);   // 256 workgroups
  dim3 block(NTHR);     // 128 threads = 4 waves
  hipLaunchKernelGGL(lcl_wmma_kernel, grid, block, 0, stream, x, w, bs, out);
}